// GraphLinear_15951508538176
// MI455X (gfx1250) — compile-verified
//
#include <hip/hip_runtime.h>
#include <hip/hip_bf16.h>

typedef float v2f __attribute__((ext_vector_type(2)));
typedef float v8f __attribute__((ext_vector_type(8)));

#define BDIM   32
#define TDIM   128
#define NNODE  32
#define FIN    256
#define FOUT   256
#define BT     (BDIM * TDIM)      // 4096
#define BT_TILE 16
#define FOUT_TILE 64

__global__ __launch_bounds__(256)
void graph_linear_wmma(const float* __restrict__ x,     // [BT, N, FIN]
                       const float* __restrict__ g,     // [N, N]
                       const int*   __restrict__ nti,   // [N]
                       const float* __restrict__ w,     // [8, FOUT, FIN]
                       const float* __restrict__ bias,  // [8, FOUT]
                       float* __restrict__ out)         // [BT, N, FOUT]
{
    // Y tile: [bt=16][n=32][o=64] fp32 = 128 KB ; g: 4 KB  (320 KB/WGP available)
    __shared__ float Ylds[BT_TILE * NNODE * FOUT_TILE];
    __shared__ float Glds[NNODE * NNODE];

    const int tid     = threadIdx.x;
    const int lane    = tid & 31;
    const int wave    = tid >> 5;
    const int lid     = lane & 15;      // row/col id within 16-half
    const int halfsel = lane >> 4;      // 0: lanes 0-15, 1: lanes 16-31

    const int bt0       = blockIdx.x * BT_TILE;
    const int fout_base = blockIdx.y * FOUT_TILE;

    // stage g into LDS (tiny)
    for (int i = tid; i < NNODE * NNODE; i += 256) Glds[i] = g[i];

    // ------------------------------------------------------------------
    // Stage 1: Y[bt,n,o] = X[bt,n,:] @ W[type(n)]^T + bias[type(n),o]
    // wave -> fout16 column (w&3), node parity (w>>2): 16 nodes per wave
    // ------------------------------------------------------------------
    const int f      = wave & 3;
    const int npar   = wave >> 2;
    const int o_base = fout_base + f * 16;

    for (int nn = 0; nn < 16; ++nn) {
        const int n = nn * 2 + npar;
        const int t = nti[n];

        // bias is per-column -> every C row gets the same value
        const float bv = bias[t * FOUT + o_base + lid];
        v8f c0 = {bv, bv, bv, bv, bv, bv, bv, bv};
        v8f c1 = {0.f, 0.f, 0.f, 0.f, 0.f, 0.f, 0.f, 0.f};

        // A (16x4, MxK): lane<16 holds K={k,k+1}, lane>=16 holds K={k+2,k+3}
        const float* ap = x + ((size_t)(bt0 + lid) * NNODE + n) * FIN + halfsel * 2;
        // B (4x16, KxN): B[k][ncol] = W[t][o_base+ncol][k]
        const float* bp = w + ((size_t)t * FOUT + o_base + lid) * FIN + halfsel * 2;

        #pragma unroll 4
        for (int k = 0; k < FIN; k += 8) {
            v2f a0 = *(const v2f*)(ap + k);
            v2f b0 = *(const v2f*)(bp + k);
            v2f a1 = *(const v2f*)(ap + k + 4);
            v2f b1 = *(const v2f*)(bp + k + 4);
            c0 = __builtin_amdgcn_wmma_f32_16x16x4_f32(false, a0, false, b0,
                                                       (short)0, c0, false, false);
            c1 = __builtin_amdgcn_wmma_f32_16x16x4_f32(false, a1, false, b1,
                                                       (short)0, c1, false, false);
        }
        c0 += c1;

        // C layout: lane col = lid, VGPR r -> row r (+8 for upper lane half)
        #pragma unroll
        for (int r = 0; r < 8; ++r) {
            const int row = r + halfsel * 8;
            Ylds[(row * NNODE + n) * FOUT_TILE + f * 16 + lid] = c0[r];
        }
    }

    __syncthreads();

    // ------------------------------------------------------------------
    // Stage 2: Z[bt,m,o] = sum_n g[m,n] * Y[bt,n,o]
    // tiles: bt(16) x mtile(2) x otile(4) = 128; 16 per wave; K=32 -> 8 WMMA
    // ------------------------------------------------------------------
    for (int tt = 0; tt < 16; ++tt) {
        const int tile = wave * 16 + tt;
        const int ot   = tile & 3;
        const int mt   = (tile >> 2) & 1;
        const int bt   = tile >> 3;

        v8f c = {0.f, 0.f, 0.f, 0.f, 0.f, 0.f, 0.f, 0.f};

        // A = g tile (16 x 4 chunk): rows of g are contiguous -> v2f LDS loads
        const float* gp = &Glds[(mt * 16 + lid) * NNODE + halfsel * 2];
        // B = Y (4 x 16): B[k][col] = Y[bt][n0+k][o0+col], n-stride = FOUT_TILE
        const float* yp = &Ylds[(bt * NNODE) * FOUT_TILE + ot * 16 + lid];

        #pragma unroll
        for (int kc = 0; kc < 8; ++kc) {
            v2f a = *(const v2f*)(gp + kc * 4);
            const int nidx = kc * 4 + halfsel * 2;
            v2f b;
            b[0] = yp[nidx * FOUT_TILE];
            b[1] = yp[(nidx + 1) * FOUT_TILE];
            c = __builtin_amdgcn_wmma_f32_16x16x4_f32(false, a, false, b,
                                                      (short)0, c, false, false);
        }

        // write Z tile: rows = m, cols = o
        float* op = out + (((size_t)(bt0 + bt) * NNODE + mt * 16) * FOUT)
                        + fout_base + ot * 16 + lid;
        #pragma unroll
        for (int r = 0; r < 8; ++r) {
            const int m = r + halfsel * 8;
            op[(size_t)m * FOUT] = c[r];
        }
    }
}

extern "C" void kernel_launch(void* const* d_in, const int* in_sizes, int n_in,
                              void* d_out, int out_size, void* d_ws, size_t ws_size,
                              hipStream_t stream) {
    (void)in_sizes; (void)n_in; (void)out_size; (void)d_ws; (void)ws_size;
    const float* x    = (const float*)d_in[0];
    const float* g    = (const float*)d_in[1];
    const int*   nti  = (const int*)d_in[2];
    const float* w    = (const float*)d_in[3];
    const float* bias = (const float*)d_in[4];
    float* out = (float*)d_out;

    dim3 grid(BT / BT_TILE, FOUT / FOUT_TILE);   // (256, 4)
    graph_linear_wmma<<<grid, 256, 0, stream>>>(x, g, nti, w, bias, out);
}